// CausalAttention_40561671143865
// MI455X (gfx1250) — compile-verified
//
#include <hip/hip_runtime.h>
#include <hip/hip_bf16.h>
#include <math.h>
#include <stdint.h>

#define B_ 8
#define S_ 2048
#define D_ 1024

typedef float v2f __attribute__((ext_vector_type(2)));
typedef float v8f __attribute__((ext_vector_type(8)));
typedef uint32_t v4u __attribute__((ext_vector_type(4)));
typedef uint32_t v8u __attribute__((ext_vector_type(8)));

static __device__ __forceinline__ v8f zero8() {
  v8f r = {0.f, 0.f, 0.f, 0.f, 0.f, 0.f, 0.f, 0.f};
  return r;
}

// D = A(16x4 f32) * B(4x16 f32) + C(16x16 f32), wave32, fp32 matrix pipe.
static __device__ __forceinline__ v8f wmma4(v2f a, v2f b, v8f c) {
  return __builtin_amdgcn_wmma_f32_16x16x4_f32(false, a, false, b, (short)0, c,
                                               false, false);
}

// ---- gfx1250 async DMA: 16B/lane global -> LDS, tracked by ASYNCcnt.
static __device__ __forceinline__ void async_b128(void* lds, const void* gptr) {
  unsigned loff = (unsigned)(uintptr_t)lds;
  asm volatile("global_load_async_to_lds_b128 %0, %1, off" ::"v"(loff),
               "v"(gptr)
               : "memory");
}
static __device__ __forceinline__ void wait_async0() {
  asm volatile("s_wait_asynccnt 0x0" ::: "memory");
}

// ---- gfx1250 Tensor Data Mover: one instruction DMAs a strided 2D tile
// into LDS, with hardware row padding matching our [rows][width+4] layouts.
// D# group1 (CDNA5 ISA); wave-uniform SGPR tuple.
static __device__ __forceinline__ v8u tdm_g1(uint32_t tensor_d0,
                                             uint32_t tensor_d1,
                                             uint32_t tile_d0, uint32_t tile_d1,
                                             uint32_t stride_elems,
                                             uint32_t pad_interval,
                                             uint32_t pad_amount) {
  v8u g1;
  g1[0] = (2u << 16)              // data_size = 4 bytes
          | (1u << 20)            // pad_enable
          | (pad_interval << 22)  // pad every 2^(i+1) DWORDs
          | (pad_amount << 25);   // pad by (a+1) DWORDs
  g1[1] = (tensor_d0 & 0xFFFFu) << 16;
  g1[2] = (tensor_d0 >> 16) | ((tensor_d1 & 0xFFFFu) << 16);
  g1[3] = (tensor_d1 >> 16) | (tile_d0 << 16);
  g1[4] = tile_d1;  // tile_dim1; tile_dim2 = 0
  g1[5] = stride_elems;  // tensor_dim0_stride[31:0]
  g1[6] = 0;
  g1[7] = 0;
  return g1;
}

static __device__ __forceinline__ void tdm_load_2d(const void* gptr, void* lds,
                                                   v8u g1) {
  uint64_t ga = (uint64_t)(uintptr_t)gptr;
  v4u g0;
  g0.x = 1u;                                 // count = 1, user descriptor
  g0.y = (uint32_t)(uintptr_t)lds;           // lds_addr
  g0.z = (uint32_t)ga;                       // global_addr[31:0]
  g0.w = (uint32_t)(ga >> 32) | (2u << 30);  // addr[56:32] | type=2 (image)
  asm volatile("tensor_load_to_lds %0, %1" ::"s"(g0), "s"(g1) : "memory");
}
static __device__ __forceinline__ void wait_tensor0() {
  __builtin_amdgcn_s_wait_tensorcnt(0);
}

static __device__ __forceinline__ void atomicMaxFloat(float* addr, float val) {
  if (val >= 0.f)
    atomicMax((int*)addr, __float_as_int(val));
  else
    atomicMin((unsigned int*)addr, __float_as_uint(val));
}

// ---------------------------------------------------------------- init M/Z
__global__ void init_mz(float* __restrict__ Mb, float* __restrict__ Zb, int n) {
  int i = blockIdx.x * blockDim.x + threadIdx.x;
  if (i < n) {
    Mb[i] = -__builtin_inff();
    Zb[i] = 0.f;
  }
}

// ---------------------------------------------------------------- QKV GEMM
// O[r, n] = sum_d X[r, d] * W[d, n];  r over B*S flattened.
// Uses the per-lane async-to-LDS path for staging.
__global__ __launch_bounds__(256) void qkv_proj(
    const float* __restrict__ X, const float* __restrict__ Wq,
    const float* __restrict__ Wk, const float* __restrict__ Wv,
    float* __restrict__ Q, float* __restrict__ K, float* __restrict__ V) {
  __shared__ float sA[128][36];
  __shared__ float sW[32][68];
  const int tid = threadIdx.x, lane = tid & 31, wave = tid >> 5;
  const int m = lane & 15, half = lane >> 4;
  const int rowBase = blockIdx.y * 128;
  const int colBase = blockIdx.x * 64;
  const float* W = (blockIdx.z == 0) ? Wq : (blockIdx.z == 1) ? Wk : Wv;
  float* O = (blockIdx.z == 0) ? Q : (blockIdx.z == 1) ? K : V;

  v8f acc[4];
#pragma unroll
  for (int t = 0; t < 4; ++t) acc[t] = zero8();

  for (int kd = 0; kd < D_; kd += 32) {
#pragma unroll
    for (int i = 0; i < 4; ++i) {  // 128x32 A tile
      int idx = tid + 256 * i, r = idx >> 3, c4 = idx & 7;
      async_b128(&sA[r][c4 * 4], &X[(size_t)(rowBase + r) * D_ + kd + c4 * 4]);
    }
#pragma unroll
    for (int i = 0; i < 2; ++i) {  // 32x64 W tile
      int idx = tid + 256 * i, r = idx >> 4, c4 = idx & 15;
      async_b128(&sW[r][c4 * 4], &W[(size_t)(kd + r) * D_ + colBase + c4 * 4]);
    }
    wait_async0();
    __syncthreads();
#pragma unroll
    for (int k4 = 0; k4 < 32; k4 += 4) {
      const int kSel = k4 + 2 * half;
      v2f a = *(const v2f*)&sA[wave * 16 + m][kSel];
#pragma unroll
      for (int ct = 0; ct < 4; ++ct) {
        v2f b;
        b.x = sW[kSel][ct * 16 + m];
        b.y = sW[kSel + 1][ct * 16 + m];
        acc[ct] = wmma4(a, b, acc[ct]);
      }
    }
    __syncthreads();
  }
#pragma unroll
  for (int ct = 0; ct < 4; ++ct)
#pragma unroll
    for (int i = 0; i < 8; ++i) {
      int r = rowBase + wave * 16 + i + 8 * half;
      int c = colBase + ct * 16 + m;
      O[(size_t)r * D_ + c] = acc[ct][i];
    }
}

// ------------------------------------------------- column max / sum passes
// mode 0: M[k] = max over valid q.  mode 1: Z[k] += exp(s - M[k]).
// Q/K tiles staged by the Tensor Data Mover (one wave issues, TENSORcnt).
__global__ __launch_bounds__(256) void col_stats(
    const float* __restrict__ Q, const float* __restrict__ K,
    float* __restrict__ Mb, float* __restrict__ Zb, int mode) {
  const int kb = blockIdx.x, qb = blockIdx.y, b = blockIdx.z;
  if (qb < kb) return;  // fully-masked tile (k > q everywhere)
  __shared__ float sQ[128][36];
  __shared__ float sK[128][36];
  const int tid = threadIdx.x, lane = tid & 31, wave = tid >> 5;
  const int m = lane & 15, half = lane >> 4;
  const size_t qBase = ((size_t)b * S_ + qb * 128) * D_;
  const size_t kBase = ((size_t)b * S_ + kb * 128) * D_;
  // 128 rows x 32 cols, row stride D; pad 4 DW after every 32 DW -> [128][36]
  const v8u g1qk = tdm_g1(D_, 128, 32, 128, D_, 4, 3);

  v8f sacc[8];
#pragma unroll
  for (int t = 0; t < 8; ++t) sacc[t] = zero8();

  for (int kd = 0; kd < D_; kd += 32) {
    if (wave == 0) {
      tdm_load_2d(&Q[qBase + kd], &sQ[0][0], g1qk);
      tdm_load_2d(&K[kBase + kd], &sK[0][0], g1qk);
      wait_tensor0();
    }
    __syncthreads();
#pragma unroll
    for (int k4 = 0; k4 < 32; k4 += 4) {
      const int kSel = k4 + 2 * half;
      v2f a = *(const v2f*)&sQ[wave * 16 + m][kSel];
#pragma unroll
      for (int ct = 0; ct < 8; ++ct) {
        v2f bf = *(const v2f*)&sK[ct * 16 + m][kSel];  // B = K^T
        sacc[ct] = wmma4(a, bf, sacc[ct]);
      }
    }
    __syncthreads();
  }

#pragma unroll
  for (int ct = 0; ct < 8; ++ct) {
    const int colg = kb * 128 + ct * 16 + m;
    if (mode == 0) {
      float red = -__builtin_inff();
#pragma unroll
      for (int i = 0; i < 8; ++i) {
        int qg = qb * 128 + wave * 16 + i + 8 * half;
        if (colg <= qg) red = fmaxf(red, sacc[ct][i]);
      }
      red = fmaxf(red, __shfl_xor(red, 16, 32));
      if (lane < 16) atomicMaxFloat(&Mb[(size_t)b * S_ + colg], red);
    } else {
      const float Mc = Mb[(size_t)b * S_ + colg];
      float red = 0.f;
#pragma unroll
      for (int i = 0; i < 8; ++i) {
        int qg = qb * 128 + wave * 16 + i + 8 * half;
        if (colg <= qg) red += __expf(sacc[ct][i] - Mc);
      }
      red += __shfl_xor(red, 16, 32);
      if (lane < 16) atomicAdd(&Zb[(size_t)b * S_ + colg], red);
    }
  }
}

// ------------------------------------------------------------- output pass
// out[q, e] = sum_{k<=q} exp(s[q,k]-M[k]) / Z[k] * v[k, e]
struct OutP1 {
  float sQ[128][36];
  float sK[128][36];
};
struct OutP2 {
  float sV[128][68];
  float sP[8][16][36];
};
union OutSmem {
  OutP1 p1;
  OutP2 p2;
};

__global__ __launch_bounds__(256) void attn_out(
    const float* __restrict__ Q, const float* __restrict__ K,
    const float* __restrict__ V, const float* __restrict__ Mb,
    const float* __restrict__ Zb, float* __restrict__ Out) {
  __shared__ OutSmem sm;
  const int eb = blockIdx.x, qb = blockIdx.y, b = blockIdx.z;
  const int tid = threadIdx.x, lane = tid & 31, wave = tid >> 5;
  const int m = lane & 15, half = lane >> 4;
  const size_t qBase = ((size_t)b * S_ + qb * 128) * D_;
  const v8u g1qk = tdm_g1(D_, 128, 32, 128, D_, 4, 3);  // [128][36]
  const v8u g1v = tdm_g1(D_, 128, 64, 128, D_, 5, 3);   // [128][68]

  v8f oacc[4];
#pragma unroll
  for (int t = 0; t < 4; ++t) oacc[t] = zero8();

  for (int kb = 0; kb <= qb; ++kb) {
    const size_t kBase = ((size_t)b * S_ + kb * 128) * D_;
    v8f sacc[8];
#pragma unroll
    for (int t = 0; t < 8; ++t) sacc[t] = zero8();

    // ---- phase 1: score tile s = q . k^T (128x128), K-loop over D
    for (int kd = 0; kd < D_; kd += 32) {
      if (wave == 0) {
        tdm_load_2d(&Q[qBase + kd], &sm.p1.sQ[0][0], g1qk);
        tdm_load_2d(&K[kBase + kd], &sm.p1.sK[0][0], g1qk);
        wait_tensor0();
      }
      __syncthreads();
#pragma unroll
      for (int k4 = 0; k4 < 32; k4 += 4) {
        const int kSel = k4 + 2 * half;
        v2f a = *(const v2f*)&sm.p1.sQ[wave * 16 + m][kSel];
#pragma unroll
        for (int ct = 0; ct < 8; ++ct) {
          v2f bf = *(const v2f*)&sm.p1.sK[ct * 16 + m][kSel];
          sacc[ct] = wmma4(a, bf, sacc[ct]);
        }
      }
      __syncthreads();
    }

    // ---- scores -> probabilities (mask, exp, column normalizer)
#pragma unroll
    for (int ct = 0; ct < 8; ++ct) {
      const int colg = kb * 128 + ct * 16 + m;
      const float Mc = Mb[(size_t)b * S_ + colg];
      const float invZ = 1.0f / Zb[(size_t)b * S_ + colg];
#pragma unroll
      for (int i = 0; i < 8; ++i) {
        int qg = qb * 128 + wave * 16 + i + 8 * half;
        float s = sacc[ct][i];
        sacc[ct][i] = (colg <= qg) ? __expf(s - Mc) * invZ : 0.0f;
      }
    }
    __syncthreads();  // retire p1 LDS before union switches to p2

    // ---- phase 2: TDM-load V tile [128k x 64e]
    if (wave == 0) {
      tdm_load_2d(&V[((size_t)b * S_ + kb * 128) * D_ + eb * 64],
                  &sm.p2.sV[0][0], g1v);
      wait_tensor0();
    }
    __syncthreads();

    // ---- P (16x128 per wave) x V (128x64), key chunks of 32 through
    //      a per-wave LDS slot that transposes C-layout -> A-layout.
#pragma unroll
    for (int kc = 0; kc < 128; kc += 32) {
#pragma unroll
      for (int cl = 0; cl < 2; ++cl) {
        int ct = (kc >> 4) + cl;
#pragma unroll
        for (int i = 0; i < 8; ++i)
          sm.p2.sP[wave][i + 8 * half][cl * 16 + m] = sacc[ct][i];
      }
#pragma unroll
      for (int k4 = 0; k4 < 32; k4 += 4) {
        const int kSel = k4 + 2 * half;
        v2f a = *(const v2f*)&sm.p2.sP[wave][m][kSel];
#pragma unroll
        for (int ct2 = 0; ct2 < 4; ++ct2) {
          v2f bf;
          bf.x = sm.p2.sV[kc + kSel][ct2 * 16 + m];
          bf.y = sm.p2.sV[kc + kSel + 1][ct2 * 16 + m];
          oacc[ct2] = wmma4(a, bf, oacc[ct2]);
        }
      }
    }
    __syncthreads();  // before next kb reuses the p1 view
  }

#pragma unroll
  for (int ct2 = 0; ct2 < 4; ++ct2)
#pragma unroll
    for (int i = 0; i < 8; ++i) {
      int qg = qb * 128 + wave * 16 + i + 8 * half;
      int eg = eb * 64 + ct2 * 16 + m;
      Out[((size_t)b * S_ + qg) * D_ + eg] = oacc[ct2][i];
    }
}

// ----------------------------------------------------------------- launch
extern "C" void kernel_launch(void* const* d_in, const int* in_sizes, int n_in,
                              void* d_out, int out_size, void* d_ws,
                              size_t ws_size, hipStream_t stream) {
  (void)in_sizes; (void)n_in; (void)out_size; (void)ws_size;
  const float* x = (const float*)d_in[0];
  const float* Wq = (const float*)d_in[1];
  const float* Wk = (const float*)d_in[2];
  const float* Wv = (const float*)d_in[3];
  float* out = (float*)d_out;

  const size_t MD = (size_t)B_ * S_ * D_;  // 16.7M floats
  float* ws = (float*)d_ws;
  float* q = ws;
  float* k = ws + MD;
  float* v = ws + 2 * MD;
  float* Mb = ws + 3 * MD;           // B*S column maxima
  float* Zb = Mb + (size_t)B_ * S_;  // B*S column exp-sums

  init_mz<<<(B_ * S_ + 255) / 256, 256, 0, stream>>>(Mb, Zb, B_ * S_);

  dim3 gProj(D_ / 64, (B_ * S_) / 128, 3);
  qkv_proj<<<gProj, 256, 0, stream>>>(x, Wq, Wk, Wv, q, k, v);

  dim3 gCol(S_ / 128, S_ / 128, B_);
  col_stats<<<gCol, 256, 0, stream>>>(q, k, Mb, Zb, 0);
  col_stats<<<gCol, 256, 0, stream>>>(q, k, Mb, Zb, 1);

  dim3 gOut(D_ / 64, S_ / 128, B_);
  attn_out<<<gOut, 256, 0, stream>>>(q, k, v, Mb, Zb, out);
}